// LGCN_Encoder_24541443129849
// MI455X (gfx1250) — compile-verified
//
#include <hip/hip_runtime.h>
#include <hip/hip_bf16.h>
#include <stdint.h>

#define USER_NUM 100000
#define ITEM_NUM 50000
#define N_TOT    (USER_NUM + ITEM_NUM)          // 150000
#define EMB      64
#define NLAYERS  3
#define CHUNK    256                             // edges per staged chunk
#define CPB      4                               // chunks per block (pipelined)

// d_out layout (floats): [user_all 6.4M][item_all 3.2M][stacked N*4*64 = 38.4M]
#define USER_FLOATS   (USER_NUM * EMB)           // 6,400,000
#define ITEM_FLOATS   (ITEM_NUM * EMB)           // 3,200,000
#define STACK_OFFSET  (USER_FLOATS + ITEM_FLOATS) // 9,600,000
#define NODE_STRIDE   (4 * EMB)                  // 256 floats per node in stacked

// Cache policy for async edge-list copies: TH=NT (bit0) so the 128MB/layer
// one-shot edge stream does not evict the L2-resident embedding slices.
#define ASYNC_CPOL 1

// ---------------------------------------------------------------------------
// CDNA5 async global->LDS copies (ASYNCcnt path), typed per clang signature.
// ---------------------------------------------------------------------------
typedef int   v2i_t __attribute__((vector_size(8)));
typedef float v4f_t __attribute__((ext_vector_type(4)));

__device__ __forceinline__ void async_copy_b64(const void* gsrc, void* lds_dst) {
#if __has_builtin(__builtin_amdgcn_global_load_async_to_lds_b64)
  __builtin_amdgcn_global_load_async_to_lds_b64(
      (v2i_t*)gsrc,
      (__attribute__((address_space(3))) v2i_t*)lds_dst, 0, ASYNC_CPOL);
#else
  uint32_t lds_off = (uint32_t)(uintptr_t)(__attribute__((address_space(3))) void*)lds_dst;
  asm volatile("global_load_async_to_lds_b64 %0, %1, off th:TH_LOAD_NT"
               :: "v"(lds_off), "v"(gsrc) : "memory");
#endif
}

__device__ __forceinline__ void async_copy_b32(const void* gsrc, void* lds_dst) {
#if __has_builtin(__builtin_amdgcn_global_load_async_to_lds_b32)
  __builtin_amdgcn_global_load_async_to_lds_b32(
      (int*)gsrc,
      (__attribute__((address_space(3))) int*)lds_dst, 0, ASYNC_CPOL);
#else
  uint32_t lds_off = (uint32_t)(uintptr_t)(__attribute__((address_space(3))) void*)lds_dst;
  asm volatile("global_load_async_to_lds_b32 %0, %1, off th:TH_LOAD_NT"
               :: "v"(lds_off), "v"(gsrc) : "memory");
#endif
}

__device__ __forceinline__ void wait_async_le0() {
#if __has_builtin(__builtin_amdgcn_s_wait_asynccnt)
  __builtin_amdgcn_s_wait_asynccnt(0);
#else
  asm volatile("s_wait_asynccnt 0" ::: "memory");
#endif
}

__device__ __forceinline__ void wait_async_le3() {
#if __has_builtin(__builtin_amdgcn_s_wait_asynccnt)
  __builtin_amdgcn_s_wait_asynccnt(3);
#else
  asm volatile("s_wait_asynccnt 3" ::: "memory");
#endif
}

// ---------------------------------------------------------------------------
// Kernel 1: layer-0 slice of stacked[] from inputs; zero slices 1..3.
// ---------------------------------------------------------------------------
__global__ __launch_bounds__(256) void lgcn_init_kernel(
    const float* __restrict__ user_emb,
    const float* __restrict__ item_emb,
    float* __restrict__ out)
{
  int idx = blockIdx.x * blockDim.x + threadIdx.x;   // over N_TOT * 16 float4s
  if (idx >= N_TOT * (EMB / 4)) return;
  int n = idx >> 4;          // node
  int q = idx & 15;          // float4 index within 64 features

  v4f_t v;
  if (n < USER_NUM) v = ((const v4f_t*)user_emb)[n * 16 + q];
  else              v = ((const v4f_t*)item_emb)[(n - USER_NUM) * 16 + q];

  v4f_t* st = (v4f_t*)(out + STACK_OFFSET) + (size_t)n * (NODE_STRIDE / 4) + q;
  const v4f_t z = {0.f, 0.f, 0.f, 0.f};
  st[0]  = v;   // layer 0
  st[16] = z;   // layer 1 (atomic dst)
  st[32] = z;   // layer 2
  st[48] = z;   // layer 3
}

// ---------------------------------------------------------------------------
// Kernel 2: one SpMM layer, double-buffered async edge staging.
// Each block: CPB chunks of CHUNK edges. Pipeline: stage chunk c+1 (3 async
// issues per wave, NT so the stream skips past L2-resident embeddings), then
// s_wait_asynccnt 3 (async loads complete in order => chunk c is landed),
// barrier, process. Each wave32 handles 32 edges/chunk: coalesced float2
// gather from the src slice, scale, native f32 atomics into the dst slice.
// ---------------------------------------------------------------------------
__global__ __launch_bounds__(256) void lgcn_spmm_kernel(
    const long long* __restrict__ rows,
    const long long* __restrict__ cols,
    const float*     __restrict__ vals,
    float* __restrict__ stacked,   // base of [N, 4, 64] region
    int src_slice,                 // (layer-1)*64
    int dst_slice,                 // layer*64
    int nnz)
{
  __shared__ long long s_rows[2][CHUNK];
  __shared__ long long s_cols[2][CHUNK];
  __shared__ float     s_vals[2][CHUNK];

  const int t    = threadIdx.x;
  const int wave = t >> 5;     // 0..7
  const int lane = t & 31;     // 2 features per lane

  const int base0 = blockIdx.x * (CHUNK * CPB);
  if (base0 >= nnz) return;

  // Stage a chunk: every wave issues exactly 3 async copies (lanes past the
  // chunk tail clamp their index) so per-wave ASYNCcnt stays uniform and the
  // "<= 3" pipeline wait is always sound.
  auto stage = [&](int cbase, int buf) {
    int cnt = nnz - cbase; if (cnt > CHUNK) cnt = CHUNK;
    int tc = (t < cnt) ? t : (cnt - 1);
    async_copy_b64(&rows[cbase + tc], &s_rows[buf][tc]);
    async_copy_b64(&cols[cbase + tc], &s_cols[buf][tc]);
    async_copy_b32(&vals[cbase + tc], &s_vals[buf][tc]);
  };

  stage(base0, 0);

  for (int c = 0; c < CPB; ++c) {
    const int cbase = base0 + c * CHUNK;
    if (cbase >= nnz) break;
    int count = nnz - cbase; if (count > CHUNK) count = CHUNK;

    const int nbase = cbase + CHUNK;
    if ((c + 1 < CPB) && (nbase < nnz)) {
      stage(nbase, (c + 1) & 1);
      wait_async_le3();          // newest 3 may remain in flight
    } else {
      wait_async_le0();
    }
    __syncthreads();             // all waves' staged data visible in LDS

    const int buf = c & 1;
    int e_begin = wave * (CHUNK / 8);
    int e_end   = e_begin + (CHUNK / 8);
    if (e_end > count) e_end = count;
    if (e_begin > count) e_begin = count;

#pragma unroll 4
    for (int e = e_begin; e < e_end; ++e) {
      // uniform-per-wave LDS reads -> broadcast
      const long long col = s_cols[buf][e];
      const long long row = s_rows[buf][e];
      const float     v   = s_vals[buf][e];

      const float2 x = *(const float2*)(stacked + (size_t)col * NODE_STRIDE
                                        + src_slice + lane * 2);
      float* d = stacked + (size_t)row * NODE_STRIDE + dst_slice + lane * 2;
      unsafeAtomicAdd(d,     x.x * v);
      unsafeAtomicAdd(d + 1, x.y * v);
    }
    __syncthreads();             // done reading buf before it is restaged
  }
}

// ---------------------------------------------------------------------------
// Kernel 3: average the 4 layer slices -> user_all / item_all sections.
// Outputs are write-once: nontemporal stores keep them out of L2.
// ---------------------------------------------------------------------------
__global__ __launch_bounds__(256) void lgcn_finalize_kernel(float* __restrict__ out)
{
  int idx = blockIdx.x * blockDim.x + threadIdx.x;   // over N_TOT * 16 float4s
  if (idx >= N_TOT * (EMB / 4)) return;
  int n = idx >> 4;
  int q = idx & 15;

  const v4f_t* st = (const v4f_t*)(out + STACK_OFFSET) + (size_t)n * (NODE_STRIDE / 4) + q;
  v4f_t a = st[0], b = st[16], c = st[32], d = st[48];
  v4f_t r = (a + b + c + d) * 0.25f;

  float* dst = (n < USER_NUM)
                 ? (out + (size_t)n * EMB)
                 : (out + USER_FLOATS + (size_t)(n - USER_NUM) * EMB);
  __builtin_nontemporal_store(r, (v4f_t*)dst + q);
}

// ---------------------------------------------------------------------------
extern "C" void kernel_launch(void* const* d_in, const int* in_sizes, int n_in,
                              void* d_out, int out_size, void* d_ws, size_t ws_size,
                              hipStream_t stream) {
  const float*     user_emb = (const float*)d_in[0];
  const float*     item_emb = (const float*)d_in[1];
  const long long* rows     = (const long long*)d_in[2];
  const long long* cols     = (const long long*)d_in[3];
  const float*     vals     = (const float*)d_in[4];
  // d_in[5] = n_layers (device scalar; cannot be read without breaking graph
  // capture). Reference constant N_LAYERS = 3.
  const int nnz = in_sizes[2];

  float* out = (float*)d_out;
  float* stacked = out + STACK_OFFSET;

  const int elem_blocks = (N_TOT * (EMB / 4) + 255) / 256;
  lgcn_init_kernel<<<elem_blocks, 256, 0, stream>>>(user_emb, item_emb, out);

  const int spmm_blocks = (nnz + CHUNK * CPB - 1) / (CHUNK * CPB);
  for (int layer = 1; layer <= NLAYERS; ++layer) {
    lgcn_spmm_kernel<<<spmm_blocks, 256, 0, stream>>>(
        rows, cols, vals, stacked, (layer - 1) * EMB, layer * EMB, nnz);
  }

  lgcn_finalize_kernel<<<elem_blocks, 256, 0, stream>>>(out);
}